// ContinuousAlphagrid_62440234549676
// MI455X (gfx1250) — compile-verified
//
#include <hip/hip_runtime.h>
#include <hip/hip_bf16.h>

// ---------------------------------------------------------------------------
// NeRF alpha-grid sampler for MI455X (gfx1250).
// Pure bandwidth kernel (235MB out / ~36MB in -> ~11.6us floor @ 23.3TB/s).
// One block per ray (16384 blocks x 512 threads, wave32 -> 16 waves/block).
//  - jitter row staged global->LDS via GLOBAL_LOAD_ASYNC_TO_LDS_B128
//    (ASYNCcnt path), consumed twice from LDS (z_vals and dists)
//  - ray data + focal are block-uniform -> s_load + SALU float pipe
//  - all 4 output streams stored NON-TEMPORAL: 235MB write-once stream must
//    not evict the 2MB density bitfield (randomly hit 8.4M times) from the
//    192MB L2
// ---------------------------------------------------------------------------

#define NRAYS     16384
#define NSAMP     512
#define STEPSZ    0.003383f
#define GRIDN     128
#define TNEAR     0.2f
#define TFAR      6.0f
#define BOUNDV    2.0f

// output layout (floats): [xyz_masked | ray_valid | z_vals | dists]
#define XYZ_CNT   ((size_t)NRAYS * NSAMP * 4)   // 33554432
#define VAL_OFF   (XYZ_CNT)
#define Z_OFF     (VAL_OFF + (size_t)NRAYS * NSAMP)
#define D_OFF     (Z_OFF   + (size_t)NRAYS * NSAMP)

typedef int v4i __attribute__((ext_vector_type(4)));
typedef float v4f __attribute__((ext_vector_type(4)));
typedef __attribute__((address_space(1))) v4i* gptr_v4i;   // global
typedef __attribute__((address_space(3))) v4i* lptr_v4i;   // LDS

__device__ __forceinline__ unsigned expand_bits3(unsigned v) {
    v = (v | (v << 16)) & 0x030000FFu;
    v = (v | (v << 8))  & 0x0300F00Fu;
    v = (v | (v << 4))  & 0x030C30C3u;
    v = (v | (v << 2))  & 0x09249249u;
    return v;
}

__global__ __launch_bounds__(NSAMP) void nerf_alphagrid_kernel(
    const float* __restrict__ rays,       // [NRAYS,6]
    const float* __restrict__ jitter,     // [NRAYS,NSAMP]
    const int*   __restrict__ bitfield,   // [2*GRIDN^3/8] (values 0..255)
    const float* __restrict__ focal,      // [1]
    float*       __restrict__ out)
{
    const int ray = blockIdx.x;
    const int s   = threadIdx.x;

    __shared__ float sj[NSAMP];
    const float* jrow = jitter + (size_t)ray * NSAMP;

    // ---- stage jitter row (2KB) into LDS via async-to-LDS ----
#if defined(__gfx1250__) && __has_builtin(__builtin_amdgcn_global_load_async_to_lds_b128)
    if (s < NSAMP / 4) {
        __builtin_amdgcn_global_load_async_to_lds_b128(
            (gptr_v4i)(jrow + 4 * s), (lptr_v4i)(&sj[4 * s]), 0, 0);
    }
  #if __has_builtin(__builtin_amdgcn_s_wait_asynccnt)
    __builtin_amdgcn_s_wait_asynccnt(0);
  #else
    asm volatile("s_wait_asynccnt 0" ::: "memory");
  #endif
#else
    sj[s] = jrow[s];
#endif
    __syncthreads();

    // ---- per-ray AABB entry t (block-uniform -> s_load + scalar f32 ALU) ----
    const float ox = rays[ray * 6 + 0];
    const float oy = rays[ray * 6 + 1];
    const float oz = rays[ray * 6 + 2];
    const float dx = rays[ray * 6 + 3];
    const float dy = rays[ray * 6 + 4];
    const float dz = rays[ray * 6 + 5];
    const float fcl = focal[0];

    const float vx = (dx == 0.0f) ? 1e-6f : dx;
    const float vy = (dy == 0.0f) ? 1e-6f : dy;
    const float vz = (dz == 0.0f) ? 1e-6f : dz;

    float tx = fminf(( BOUNDV - ox) / vx, (-BOUNDV - ox) / vx);
    float ty = fminf(( BOUNDV - oy) / vy, (-BOUNDV - oy) / vy);
    float tz = fminf(( BOUNDV - oz) / vz, (-BOUNDV - oz) / vz);
    float tmin = fmaxf(fmaxf(tx, ty), tz);
    tmin = fminf(fmaxf(tmin, TNEAR), TFAR);

    // ---- per-sample ----
    const float jit    = sj[s];
    const float interp = tmin + STEPSZ * ((float)s + jit);

    const float px = ox + dx * interp;
    const float py = oy + dy * interp;
    const float pz = oz + dz * interp;

    const bool inb = (px >= -BOUNDV) & (px <= BOUNDV) &
                     (py >= -BOUNDV) & (py <= BOUNDV) &
                     (pz >= -BOUNDV) & (pz <= BOUNDV);

    // bound0 = min(1, BOUND) = 1 ; divide by (1 - 1/128) == * 128/127
    const float scl = (float)GRIDN / (float)(GRIDN - 1);
    float nx = fminf(fmaxf(px * scl, -1.0f), 1.0f);
    float ny = fminf(fmaxf(py * scl, -1.0f), 1.0f);
    float nz = fminf(fmaxf(pz * scl, -1.0f), 1.0f);

    const unsigned cx = (unsigned)(int)((nx + 1.0f) * 0.5f * (float)(GRIDN - 1));
    const unsigned cy = (unsigned)(int)((ny + 1.0f) * 0.5f * (float)(GRIDN - 1));
    const unsigned cz = (unsigned)(int)((nz + 1.0f) * 0.5f * (float)(GRIDN - 1));

    const unsigned midx = expand_bits3(cx) | (expand_bits3(cy) << 1) |
                          (expand_bits3(cz) << 2);

    const int   word  = bitfield[midx >> 3];           // random hit, L2-resident
    const bool  alpha = ((word >> (midx & 7u)) & 1) != 0;
    const bool  valid = inb & alpha;
    const float vm    = valid ? 1.0f : 0.0f;

    float dist = 0.0f;
    if (s < NSAMP - 1) {
        const float jn = sj[s + 1];
        const float interp_n = tmin + STEPSZ * ((float)(s + 1) + jn);
        dist = interp_n - interp;
    }

    // ---- coalesced NON-TEMPORAL stores (write-once 235MB stream) ----
    const size_t gi = (size_t)ray * NSAMP + (size_t)s;

    v4f xm;
    xm.x = px * vm;
    xm.y = py * vm;
    xm.z = pz * vm;
    xm.w = (interp / fcl) * vm;
    __builtin_nontemporal_store(xm, reinterpret_cast<v4f*>(out) + gi);
    __builtin_nontemporal_store(vm,     out + VAL_OFF + gi);
    __builtin_nontemporal_store(interp, out + Z_OFF   + gi);
    __builtin_nontemporal_store(dist,   out + D_OFF   + gi);
}

extern "C" void kernel_launch(void* const* d_in, const int* in_sizes, int n_in,
                              void* d_out, int out_size, void* d_ws, size_t ws_size,
                              hipStream_t stream) {
    const float* rays     = (const float*)d_in[0];   // [16384,6]
    const float* jitter   = (const float*)d_in[1];   // [16384,512]
    /* d_in[2] density_grid: dead input, never read by reference outputs */
    const int*   bitfield = (const int*)d_in[3];     // [524288]
    const float* focal    = (const float*)d_in[4];   // [1]
    float*       out      = (float*)d_out;

    nerf_alphagrid_kernel<<<NRAYS, NSAMP, 0, stream>>>(rays, jitter, bitfield,
                                                       focal, out);
}